// ChamferDistance_78761110274577
// MI455X (gfx1250) — compile-verified
//
#include <hip/hip_runtime.h>
#include <hip/hip_bf16.h>

typedef __attribute__((ext_vector_type(2))) float v2f;
typedef __attribute__((ext_vector_type(8))) float v8f;

#define TILES_PER_WAVE 2   // 16-row tiles per wave (32 query points)
#define WAVES_PER_BLOCK 8  // 256 threads / wave32

// One direction of Chamfer NN search:
// for each point in X [B,N,3], min squared distance + argmin over Y [B,M,3].
// d(m,n) = |x|^2 - 2 x.y + |y|^2 ; the -2x.y + |y|^2 part is computed as a
// K=4 f32 WMMA with A row = (x0,x1,x2,1), B col = (-2y0,-2y1,-2y2,|y|^2),
// C = inline 0. |x|^2 is row-constant -> added once in the epilogue.
__global__ __launch_bounds__(256) void chamfer_nn_dir_kernel(
    const float* __restrict__ X,   // [B, N, 3] queries
    const float* __restrict__ Y,   // [B, M, 3] references
    float* __restrict__ outD,      // [B, N]
    int*   __restrict__ outI,      // [B, N]
    int N, int M)
{
    const int lane = threadIdx.x & 31;
    const int wave = threadIdx.x >> 5;
    const int waveGlobal = blockIdx.x * WAVES_PER_BLOCK + wave;
    const int tilesPerBatch = N >> 4;

    const int tile0 = waveGlobal * TILES_PER_WAVE;       // first 16-row tile
    const int batch = tile0 / tilesPerBatch;
    const int rowTile0 = tile0 - batch * tilesPerBatch;  // tile index in batch

    const float* __restrict__ xb = X + (size_t)batch * N * 3;
    const float* __restrict__ yb = Y + (size_t)batch * M * 3;

    const int  n16 = lane & 15;
    const bool hi  = lane >= 16;

    // ---- Build A operands (one per row tile), loop-invariant. ----
    // A layout (16x4 f32, MxK): lanes 0-15 hold M=lane, VGPR0=K0, VGPR1=K1;
    // lanes 16-31 hold K2, K3 for the same M=lane-16.
    v2f a[TILES_PER_WAVE];
#pragma unroll
    for (int t = 0; t < TILES_PER_WAVE; ++t) {
        const int row = ((rowTile0 + t) << 4) + n16;
        const float* q = xb + row * 3;
        const float q0 = q[0], q1 = q[1], q2 = q[2];
        a[t].x = hi ? q2   : q0;   // K2 : K0
        a[t].y = hi ? 1.0f : q1;   // K3 : K1
    }

    // Running min / argmin of q(m,n) = -2x.y + |y|^2 per C/D element.
    float rmin[TILES_PER_WAVE][8];
    int   ridx[TILES_PER_WAVE][8];
#pragma unroll
    for (int t = 0; t < TILES_PER_WAVE; ++t)
#pragma unroll
        for (int r = 0; r < 8; ++r) { rmin[t][r] = 3.4e38f; ridx[t][r] = 0; }

    // ---- Sweep all reference tiles (16 points each). ----
    for (int c0 = 0; c0 < M; c0 += 16) {
        const int colIdx = c0 + n16;             // this lane's column index
        const float* p = yb + (size_t)colIdx * 3;
        const float y0 = p[0], y1 = p[1], y2c = p[2];
        const float s  = fmaf(y0, y0, fmaf(y1, y1, y2c * y2c));  // |y|^2
        const float m0 = -2.0f * y0, m1 = -2.0f * y1, m2 = -2.0f * y2c;

        // B layout (4x16 f32, KxN): lanes 0-15 hold N=lane with VGPR0=K0,
        // VGPR1=K1; lanes 16-31 hold K2, K3 for N=lane-16.
        v2f b;
        b.x = hi ? m2 : m0;   // K2 : K0
        b.y = hi ? s  : m1;   // K3 : K1

#pragma unroll
        for (int t = 0; t < TILES_PER_WAVE; ++t) {
            v8f cz = {};  // inline constant 0 for the C operand
            v8f d = __builtin_amdgcn_wmma_f32_16x16x4_f32(
                /*neg_a=*/false, a[t], /*neg_b=*/false, b,
                /*c_mod=*/(short)0, cz, /*reuse_a=*/false, /*reuse_b=*/false);
#pragma unroll
            for (int r = 0; r < 8; ++r) {
                const float dv = d[r];
                if (dv < rmin[t][r]) { rmin[t][r] = dv; ridx[t][r] = colIdx; }
            }
        }
    }

    // ---- Epilogue: reduce across the 16 lanes of each half (lanes 0-15
    // carry rows r, lanes 16-31 carry rows r+8), tie-break on lowest index
    // to match jnp.argmin's first-occurrence semantics. ----
#pragma unroll
    for (int t = 0; t < TILES_PER_WAVE; ++t) {
        const int rowBase = ((rowTile0 + t) << 4);
#pragma unroll
        for (int r = 0; r < 8; ++r) {
            float dv = rmin[t][r];
            int   iv = ridx[t][r];
#pragma unroll
            for (int off = 8; off >= 1; off >>= 1) {  // stays within 16-lane half
                const float od = __shfl_xor(dv, off, 32);
                const int   oi = __shfl_xor(iv, off, 32);
                if (od < dv || (od == dv && oi < iv)) { dv = od; iv = oi; }
            }
            if (n16 == 0) {   // lane 0 -> row rowBase+r ; lane 16 -> rowBase+8+r
                const int row = rowBase + r + (hi ? 8 : 0);
                const float* q = xb + (size_t)row * 3;
                const float xs = fmaf(q[0], q[0], fmaf(q[1], q[1], q[2] * q[2]));
                outD[(size_t)batch * N + row] = dv + xs;  // add |x|^2 back
                outI[(size_t)batch * N + row] = iv;
            }
        }
    }
}

extern "C" void kernel_launch(void* const* d_in, const int* in_sizes, int n_in,
                              void* d_out, int out_size, void* d_ws, size_t ws_size,
                              hipStream_t stream) {
    (void)n_in; (void)in_sizes; (void)d_ws; (void)ws_size; (void)out_size;

    const int B = 8, N = 16384, M = 16384;

    const float* xyz1 = (const float*)d_in[0];
    const float* xyz2 = (const float*)d_in[1];

    float* dist1 = (float*)d_out;             // [B, N]
    float* dist2 = dist1 + (size_t)B * N;     // [B, M]
    int*   idx1  = (int*)(dist2 + (size_t)B * M);  // [B, N] int32 bits
    int*   idx2  = idx1 + (size_t)B * N;           // [B, M]

    // waves per direction = B * (N/16) / TILES_PER_WAVE
    const int totalWaves = B * (N >> 4) / TILES_PER_WAVE;
    const int blocks = totalWaves / WAVES_PER_BLOCK;

    chamfer_nn_dir_kernel<<<blocks, WAVES_PER_BLOCK * 32, 0, stream>>>(
        xyz1, xyz2, dist1, idx1, N, M);
    chamfer_nn_dir_kernel<<<blocks, WAVES_PER_BLOCK * 32, 0, stream>>>(
        xyz2, xyz1, dist2, idx2, M, N);
}